// ConvRouting_24747601560077
// MI455X (gfx1250) — compile-verified
//
#include <hip/hip_runtime.h>
#include <hip/hip_bf16.h>

typedef __attribute__((ext_vector_type(2))) float v2f;
typedef __attribute__((ext_vector_type(8))) float v8f;

#define BATCHES 16
#define NPAIRS  8               // two batches per WMMA fragment
#define HW      4096            // 64*64
#define NGRP    8
#define CPG     64              // channels per group
#define CG      512             // NGRP*CPG
#define KTOT    (HW * CPG)      // 262144 contraction length per batch
#define GRAM_BLOCKS_Y    32
#define WAVES_PER_BLOCK  8
#define WAVES_PER_PAIR   (GRAM_BLOCKS_Y * WAVES_PER_BLOCK)   // 256
#define K_PER_WAVE       (KTOT / WAVES_PER_PAIR)             // 1024

// ---------------------------------------------------------------------------
// Kernel 1: Gram matrices via V_WMMA_F32_16X16X4_F32, TWO batches per wave.
// Fragment row/col m: m<8 -> batch (2p) group m ; m>=8 -> batch (2p+1) group m-8.
// Accumulator top-left 8x8 = Gram(batch even), bottom-right 8x8 = Gram(batch odd);
// off-diagonal cross-batch blocks are ignored. No dead lanes -> no exec-mask
// zeroing. Loop is software-pipelined: next b128 load issues before the WMMA.
// ---------------------------------------------------------------------------
__global__ void __launch_bounds__(256) gram_kernel(const float* __restrict__ x,
                                                   float* __restrict__ P) {
  const int pair = blockIdx.x;                       // 0..7
  const int lane = threadIdx.x & 31;
  const int wave = threadIdx.x >> 5;
  const int wg   = blockIdx.y * WAVES_PER_BLOCK + wave;  // 0..255 per pair

  const int  m  = lane & 15;                         // fragment row/col
  const int  g  = m & 7;                             // group within batch
  const int  bb = pair * 2 + (m >> 3);               // which batch this row is
  const bool hi = lane >= 16;                        // K-half selector

  const float* xb = x + (size_t)bb * HW * CG;
  const size_t k0 = (size_t)wg * K_PER_WAVE;

  // address of x[bb, hw(k), g, c(k)] for a 4-aligned k (all 4 k share one hw row)
  auto addr = [&](size_t kb) {
    return xb + (kb >> 6) * CG + (size_t)g * CPG + (kb & 63);
  };

  float4 q = *reinterpret_cast<const float4*>(addr(k0));
  v8f acc = {};

  for (int kk = 0; kk < K_PER_WAVE; kk += 4) {
    // preload next chunk (wrap to k0 on last trip: valid address, value unused)
    const size_t knext = (kk + 4 < K_PER_WAVE) ? (k0 + kk + 4) : k0;
    const float* pn = addr(knext);
    __builtin_prefetch(pn + 2048, 0, 1);             // global_prefetch_b8, 8KB ahead
    float4 qn = *reinterpret_cast<const float4*>(pn);

    // A fragment (16x4): lanes 0-15 carry K0/K1, lanes 16-31 carry K2/K3
    const float a0 = hi ? q.z : q.x;
    const float a1 = hi ? q.w : q.y;
    // B fragment (4x16): VGPR0 = rows K0|K1 across lane halves, VGPR1 = rows K2|K3
    const float b0 = hi ? q.y : q.x;
    const float b1 = hi ? q.w : q.z;

    v2f A = {a0, a1};
    v2f B = {b0, b1};
    acc = __builtin_amdgcn_wmma_f32_16x16x4_f32(false, A, false, B,
                                                (short)0, acc, false, false);
    q = qn;
  }

  // Extract both 8x8 Gram tiles (C/D layout: VGPR j, lanes 0-15 -> M=j,N=lane;
  // lanes 16-31 -> M=j+8, N=lane-16).
  if (lane < 8) {                                    // batch even: rows 0-7, cols 0-7
    float* dst = P + ((size_t)((pair * 2 + 0) * WAVES_PER_PAIR + wg) * 64) + lane;
#pragma unroll
    for (int j = 0; j < 8; ++j) dst[j * 8] = acc[j];
  } else if (lane >= 24) {                           // batch odd: rows 8-15, cols 8-15
    float* dst = P + ((size_t)((pair * 2 + 1) * WAVES_PER_PAIR + wg) * 64) + (lane - 24);
#pragma unroll
    for (int j = 0; j < 8; ++j) dst[j * 8] = acc[j];
  }
}

// ---------------------------------------------------------------------------
// Kernel 2: deterministic fixed-order reduction of per-wave partials into
// S[b*64 + g*8 + g'].
// ---------------------------------------------------------------------------
__global__ void __launch_bounds__(256) reduce_kernel(const float* __restrict__ P,
                                                     float* __restrict__ S) {
  const int i = blockIdx.x * 256 + threadIdx.x;   // 0..1023
  const int b = i >> 6;
  const int e = i & 63;
  float s = 0.0f;
  for (int w = 0; w < WAVES_PER_PAIR; ++w)
    s += P[((size_t)(b * WAVES_PER_PAIR + w) * 64) + e];
  S[i] = s;
}

// ---------------------------------------------------------------------------
// Kernel 3: recompute the 2-step routing recurrence from S (uniform -> scalar),
// then stream v = sum_g alpha_g * x_g with float4 traffic. x (134MB) fits in
// the 192MB L2, so this second pass mostly hits L2.
// ---------------------------------------------------------------------------
__global__ void __launch_bounds__(256) route_out_kernel(const float* __restrict__ x,
                                                        const float* __restrict__ S,
                                                        float* __restrict__ out) {
  const int b = blockIdx.x;
  const float* Sb = S + b * 64;

  float beta[8], alpha[8];
#pragma unroll
  for (int g = 0; g < 8; ++g) {             // iter 0: alpha == 0.5
    float s = 0.0f;
#pragma unroll
    for (int gp = 0; gp < 8; ++gp) s += Sb[g * 8 + gp];
    beta[g] = 0.5f * s;
  }
#pragma unroll
  for (int g = 0; g < 8; ++g) alpha[g] = 1.0f / (1.0f + __expf(-beta[g]));
#pragma unroll
  for (int g = 0; g < 8; ++g) {             // iter 1 agreement
    float s = beta[g];
#pragma unroll
    for (int gp = 0; gp < 8; ++gp) s += alpha[gp] * Sb[gp * 8 + g];
    beta[g] = s;
  }
#pragma unroll
  for (int g = 0; g < 8; ++g) alpha[g] = 1.0f / (1.0f + __expf(-beta[g]));

  // 16 hw positions per block, 16 float4 channel-chunks each.
  const int t  = threadIdx.x;
  const int c4 = t & 15;
  const int hw = blockIdx.y * 16 + (t >> 4);

  const float4* xp = reinterpret_cast<const float4*>(x + ((size_t)b * HW + hw) * CG) + c4;
  float4 acc; acc.x = 0.0f; acc.y = 0.0f; acc.z = 0.0f; acc.w = 0.0f;
#pragma unroll
  for (int g = 0; g < 8; ++g) {
    float4 v = xp[g * 16];                   // group stride = 64 floats
    acc.x += alpha[g] * v.x;
    acc.y += alpha[g] * v.y;
    acc.z += alpha[g] * v.z;
    acc.w += alpha[g] * v.w;
  }
  *(reinterpret_cast<float4*>(out + ((size_t)b * HW + hw) * CPG) + c4) = acc;
}

// ---------------------------------------------------------------------------
extern "C" void kernel_launch(void* const* d_in, const int* in_sizes, int n_in,
                              void* d_out, int out_size, void* d_ws, size_t ws_size,
                              hipStream_t stream) {
  const float* x = reinterpret_cast<const float*>(d_in[0]);
  float* out = reinterpret_cast<float*>(d_out);

  // workspace: P = 16 batches * 256 wave-partials * 64 floats (1 MB), then S (4 KB)
  float* P = reinterpret_cast<float*>(d_ws);
  float* S = P + (size_t)BATCHES * WAVES_PER_PAIR * 64;

  gram_kernel<<<dim3(NPAIRS, GRAM_BLOCKS_Y), 256, 0, stream>>>(x, P);
  reduce_kernel<<<4, 256, 0, stream>>>(P, S);
  route_out_kernel<<<dim3(BATCHES, HW / 16), 256, 0, stream>>>(x, S, out);
}